// GNNEncoder_86071144611930
// MI455X (gfx1250) — compile-verified
//
#include <hip/hip_runtime.h>
#include <math.h>

// Problem constants (match reference)
#define NCELL   65536            // B*NC
#define NTIS    4096             // B*NT
#define HDIM    256
#define FIN_    514
#define DOUT_   128
#define BATCH   16
#define NC_     4096
#define NT_     256
#define DEG_    16
#define NE_CELL (BATCH*NC_*DEG_) // 1048576
#define NE_TIS  (BATCH*NT_*DEG_) // 65536
#define EPS_BN   1e-5f
#define EPS_NORM 1e-12f

typedef __attribute__((ext_vector_type(2))) float v2f;
typedef __attribute__((ext_vector_type(8))) float v8f;

__device__ __forceinline__ v8f wmma_f32(v2f a, v2f b, v8f c) {
    return __builtin_amdgcn_wmma_f32_16x16x4_f32(
        /*neg_a=*/false, a, /*neg_b=*/false, b,
        /*c_mod=*/(short)0, c, /*reuse_a=*/false, /*reuse_b=*/false);
}

// ---------------------------------------------------------------------------
// f32 WMMA GEMM:  C[b] = op(A[b]) @ B[b]  (+ rowscale, + bias)
//   A element (m,k) at A + b*sAb + m*sAm + k*sAk   (CONTIG_K => sAk==1)
//   B element (k,n) at B + b*sBb + k*N + n         (row-major, ld == N)
//   C element (m,n) at C + b*sCb + m*N + n
// One 16x64 output strip per wave32 (4 WMMA tiles share one A fetch).
// Main K loop is guard-free; K%4 remainder handled once at the end.
// ---------------------------------------------------------------------------
template <bool CONTIG_K>
__global__ void __launch_bounds__(256)
gemm_wmma_f32(const float* __restrict__ A, long long sAm, long long sAk, long long sAb,
              const float* __restrict__ B, long long sBb,
              float* __restrict__ C, long long sCb,
              int M, int N, int K,
              const float* __restrict__ rowscale,
              const float* __restrict__ bias)
{
    const int lane = threadIdx.x & 31;
    const int wave = threadIdx.x >> 5;
    const int tilesN = N >> 6;                       // 64-wide strips
    const long long tiles = (long long)(M >> 4) * tilesN;
    const long long tileId = (long long)blockIdx.x * 8 + wave;
    if (tileId >= tiles) return;                     // wave-uniform
    const int tm = (int)(tileId / tilesN);
    const int tn = (int)(tileId % tilesN);
    const int b  = blockIdx.y;

    const int half = lane >> 4;                      // 0: K {0,1}, 1: K {2,3}
    const int lm   = lane & 15;
    const int col0 = tn * 64 + lm;                   // first of 4 columns

    // per-lane streaming pointers
    const float* Ap = A + (long long)b * sAb
                        + (long long)(tm * 16 + lm) * sAm
                        + (long long)(half * 2) * sAk;
    const float* Bp = B + (long long)b * sBb
                        + (long long)(half * 2) * N + col0;
    float* Cp = C + (long long)b * sCb;

    v8f acc0 = {}, acc1 = {}, acc2 = {}, acc3 = {};
    const int Kmain = K & ~3;

    #pragma unroll 2
    for (int k0 = 0; k0 < Kmain; k0 += 4) {
        v2f av;
        if (CONTIG_K) {
            const float2 t = *(const float2*)Ap;     // b64 load, 8B aligned
            av[0] = t.x; av[1] = t.y;
        } else {
            av[0] = Ap[0]; av[1] = Ap[sAk];
        }
        v2f b0, b1, b2, b3;
        b0[0] = Bp[0];  b0[1] = Bp[N];
        b1[0] = Bp[16]; b1[1] = Bp[N + 16];
        b2[0] = Bp[32]; b2[1] = Bp[N + 32];
        b3[0] = Bp[48]; b3[1] = Bp[N + 48];
        acc0 = wmma_f32(av, b0, acc0);
        acc1 = wmma_f32(av, b1, acc1);
        acc2 = wmma_f32(av, b2, acc2);
        acc3 = wmma_f32(av, b3, acc3);
        Ap += 4 * sAk;
        Bp += (long long)4 * N;
    }

    if (Kmain < K) {                                 // K%4 tail (once, guarded)
        const int ka = Kmain + half * 2;
        const bool v0 = (ka < K), v1 = (ka + 1 < K);
        v2f av;
        av[0] = v0 ? Ap[0]   : 0.f;
        av[1] = v1 ? Ap[sAk] : 0.f;
        v2f b0, b1, b2, b3;
        b0[0] = v0 ? Bp[0]      : 0.f;  b0[1] = v1 ? Bp[N]      : 0.f;
        b1[0] = v0 ? Bp[16]     : 0.f;  b1[1] = v1 ? Bp[N + 16] : 0.f;
        b2[0] = v0 ? Bp[32]     : 0.f;  b2[1] = v1 ? Bp[N + 32] : 0.f;
        b3[0] = v0 ? Bp[48]     : 0.f;  b3[1] = v1 ? Bp[N + 48] : 0.f;
        acc0 = wmma_f32(av, b0, acc0);
        acc1 = wmma_f32(av, b1, acc1);
        acc2 = wmma_f32(av, b2, acc2);
        acc3 = wmma_f32(av, b3, acc3);
    }

    // C/D layout: VGPR r holds M = tm*16 + r + half*8, N = col
    const float bb0 = bias ? bias[col0]      : 0.f;
    const float bb1 = bias ? bias[col0 + 16] : 0.f;
    const float bb2 = bias ? bias[col0 + 32] : 0.f;
    const float bb3 = bias ? bias[col0 + 48] : 0.f;
    #pragma unroll
    for (int r = 0; r < 8; ++r) {
        const int m = tm * 16 + r + half * 8;
        const float sc = rowscale ? rowscale[m] : 1.f;
        float* crow = Cp + (long long)m * N + col0;
        crow[0]  = acc0[r] * sc + bb0;
        crow[16] = acc1[r] * sc + bb1;
        crow[32] = acc2[r] * sc + bb2;
        crow[48] = acc3[r] * sc + bb3;
    }
}

// ---------------------------------------------------------------------------
// Elementwise / graph kernels
// ---------------------------------------------------------------------------
__global__ void zero_f32(float* __restrict__ p, long long n) {
    long long i = (long long)blockIdx.x * blockDim.x + threadIdx.x;
    if (i < n) p[i] = 0.f;
}

__global__ void deg_kernel(const int* __restrict__ src, const int* __restrict__ dst,
                           float* __restrict__ dout, float* __restrict__ din, int ne) {
    int i = blockIdx.x * blockDim.x + threadIdx.x;
    if (i < ne) {
        atomicAdd(&dout[src[i]], 1.f);
        atomicAdd(&din[dst[i]], 1.f);
    }
}

__global__ void invsqrt_kernel(float* __restrict__ d, int n) {
    int i = blockIdx.x * blockDim.x + threadIdx.x;
    if (i < n) {
        float v = d[i];
        d[i] = (v > 0.f) ? rsqrtf(fmaxf(v, 1.f)) : 0.f;
    }
}

// agg[dst[e], f] += xw[src[e], f]  — feature index fastest for coalescing (H=256)
__global__ void scatter_kernel(const int* __restrict__ src, const int* __restrict__ dst,
                               const float* __restrict__ xw, float* __restrict__ agg,
                               long long total) {
    long long i = (long long)blockIdx.x * blockDim.x + threadIdx.x;
    if (i < total) {
        const int f = (int)(i & (HDIM - 1));
        const int e = (int)(i >> 8);
        atomicAdd(&agg[(long long)dst[e] * HDIM + f],
                  xw[(long long)src[e] * HDIM + f]);
    }
}

// Column sum / sumsq of t = agg*ni + bias  (BN stats pass). blockDim == HDIM.
__global__ void __launch_bounds__(HDIM)
stats_kernel(const float* __restrict__ agg, const float* __restrict__ ni,
             const float* __restrict__ bias,
             float* __restrict__ colsum, float* __restrict__ colsq,
             int rowsPerBlock) {
    const int j  = threadIdx.x;
    const int r0 = blockIdx.x * rowsPerBlock;
    const float bj = bias[j];
    float s = 0.f, ss = 0.f;
    for (int r = r0; r < r0 + rowsPerBlock; ++r) {
        float t = agg[(long long)r * HDIM + j] * ni[r] + bj;
        s += t; ss += t * t;
    }
    atomicAdd(&colsum[j], s);
    atomicAdd(&colsq[j], ss);
}

// out = BN(agg*ni + bias) with gamma/beta
__global__ void apply_bn_kernel(const float* __restrict__ agg, const float* __restrict__ ni,
                                const float* __restrict__ bias,
                                const float* __restrict__ colsum, const float* __restrict__ colsq,
                                const float* __restrict__ g, const float* __restrict__ beta,
                                float* __restrict__ out, int nrows) {
    long long i = (long long)blockIdx.x * blockDim.x + threadIdx.x;
    const long long total = (long long)nrows * HDIM;
    if (i < total) {
        const int j = (int)(i & (HDIM - 1));
        const int r = (int)(i >> 8);
        const float nf = (float)nrows;
        const float mean = colsum[j] / nf;
        const float var  = colsq[j] / nf - mean * mean;
        const float t = agg[i] * ni[r] + bias[j];
        out[i] = (t - mean) * rsqrtf(var + EPS_BN) * g[j] + beta[j];
    }
}

// xt = [h_agg | tissue_feat]  -> [NTIS, HDIM+FIN]
__global__ void concat_kernel(const float* __restrict__ hagg,
                              const float* __restrict__ tfeat,
                              float* __restrict__ xt) {
    const int W = HDIM + FIN_;
    const long long total = (long long)NTIS * W;
    long long i = (long long)blockIdx.x * blockDim.x + threadIdx.x;
    if (i < total) {
        const int r = (int)(i / W);
        const int c = (int)(i % W);
        xt[i] = (c < HDIM) ? hagg[(long long)r * HDIM + c]
                           : tfeat[(long long)r * FIN_ + (c - HDIM)];
    }
}

// Mean-pool over NT rows per graph, then L2-normalize across DOUT.
__global__ void __launch_bounds__(DOUT_)
pool_norm_kernel(const float* __restrict__ y, float* __restrict__ out) {
    __shared__ float red[DOUT_];
    const int b = blockIdx.x;
    const int d = threadIdx.x;
    float s = 0.f;
    for (int t = 0; t < NT_; ++t)
        s += y[(long long)(b * NT_ + t) * DOUT_ + d];
    s /= (float)NT_;
    red[d] = s * s;
    __syncthreads();
    for (int off = DOUT_ / 2; off > 0; off >>= 1) {
        if (d < off) red[d] += red[d + off];
        __syncthreads();
    }
    const float norm = fmaxf(sqrtf(red[0]), EPS_NORM);
    out[b * DOUT_ + d] = s / norm;
}

// ---------------------------------------------------------------------------
extern "C" void kernel_launch(void* const* d_in, const int* in_sizes, int n_in,
                              void* d_out, int out_size, void* d_ws, size_t ws_size,
                              hipStream_t stream) {
    (void)in_sizes; (void)n_in; (void)out_size; (void)ws_size;
    const float* cell_feat   = (const float*)d_in[0];
    const float* tissue_feat = (const float*)d_in[1];
    const float* assign      = (const float*)d_in[2];
    const int*   cell_src    = (const int*)d_in[3];
    const int*   cell_dst    = (const int*)d_in[4];
    const int*   tis_src     = (const int*)d_in[5];
    const int*   tis_dst     = (const int*)d_in[6];
    const float* cell_W0     = (const float*)d_in[7];
    const float* cell_b0     = (const float*)d_in[8];
    const float* cell_Ws     = (const float*)d_in[9];
    const float* cell_bs     = (const float*)d_in[10];
    const float* cell_bn_g   = (const float*)d_in[11];
    const float* cell_bn_b   = (const float*)d_in[12];
    const float* tis_W0      = (const float*)d_in[13];
    const float* tis_b0      = (const float*)d_in[14];
    const float* tis_Ws      = (const float*)d_in[15];
    const float* tis_bs      = (const float*)d_in[16];
    const float* tis_bn_g    = (const float*)d_in[17];
    const float* tis_bn_b    = (const float*)d_in[18];
    const float* lin_W       = (const float*)d_in[19];
    const float* lin_b       = (const float*)d_in[20];
    float* out = (float*)d_out;

    // Workspace carve-up (all f32)
    float* w = (float*)d_ws;
    float* buf0 = w; w += (long long)NCELL * HDIM;   // cell h (rotating)
    float* buf1 = w; w += (long long)NCELL * HDIM;   // cell x@W
    float* buf2 = w; w += (long long)NCELL * HDIM;   // cell agg
    float* cns  = w; w += NCELL;
    float* cni  = w; w += NCELL;
    float* tns  = w; w += NTIS;
    float* tni  = w; w += NTIS;
    float* colsum = w; w += HDIM;
    float* colsq  = w; w += HDIM;
    float* hagg = w; w += (long long)NTIS * HDIM;
    float* xt   = w; w += (long long)NTIS * (HDIM + FIN_);
    float* tb0  = w; w += (long long)NTIS * HDIM;
    float* tb1  = w; w += (long long)NTIS * HDIM;
    float* tb2  = w; w += (long long)NTIS * HDIM;
    float* yv   = w; w += (long long)NTIS * DOUT_;

    auto zero = [&](float* p, long long n) {
        zero_f32<<<(unsigned)((n + 255) / 256), 256, 0, stream>>>(p, n);
    };
    auto gemm = [&](bool contig,
                    const float* A, long long sAm, long long sAk, long long sAb,
                    const float* Bm, long long sBb, float* Cm, long long sCb,
                    int M, int N, int K,
                    const float* rs, const float* bias, int nb) {
        long long tiles = (long long)(M / 16) * (N / 64);
        dim3 grid((unsigned)((tiles + 7) / 8), (unsigned)nb);
        if (contig)
            gemm_wmma_f32<true><<<grid, 256, 0, stream>>>(A, sAm, sAk, sAb, Bm, sBb,
                                                          Cm, sCb, M, N, K, rs, bias);
        else
            gemm_wmma_f32<false><<<grid, 256, 0, stream>>>(A, sAm, sAk, sAb, Bm, sBb,
                                                           Cm, sCb, M, N, K, rs, bias);
    };

    // --- degree normalizers: ns = D_out^{-1/2}, ni = D_in^{-1/2} ---
    zero(cns, NCELL); zero(cni, NCELL);
    deg_kernel<<<(NE_CELL + 255) / 256, 256, 0, stream>>>(cell_src, cell_dst, cns, cni, NE_CELL);
    invsqrt_kernel<<<(NCELL + 255) / 256, 256, 0, stream>>>(cns, NCELL);
    invsqrt_kernel<<<(NCELL + 255) / 256, 256, 0, stream>>>(cni, NCELL);
    zero(tns, NTIS); zero(tni, NTIS);
    deg_kernel<<<(NE_TIS + 255) / 256, 256, 0, stream>>>(tis_src, tis_dst, tns, tni, NE_TIS);
    invsqrt_kernel<<<(NTIS + 255) / 256, 256, 0, stream>>>(tns, NTIS);
    invsqrt_kernel<<<(NTIS + 255) / 256, 256, 0, stream>>>(tni, NTIS);

    // --- cell GCN stack: h = BN( (X@W * ns)[src] scattered to dst * ni + b ) ---
    {
        const float* X  = cell_feat; int K = FIN_;
        const float* Wt = cell_W0;   const float* bg = cell_b0;
        for (int layer = 0; layer < 3; ++layer) {
            gemm(true, X, K, 1, 0, Wt, 0, buf1, 0, NCELL, HDIM, K, cns, nullptr, 1);
            zero(buf2, (long long)NCELL * HDIM);
            long long tot = (long long)NE_CELL * HDIM;
            scatter_kernel<<<(unsigned)((tot + 255) / 256), 256, 0, stream>>>(
                cell_src, cell_dst, buf1, buf2, tot);
            zero(colsum, HDIM); zero(colsq, HDIM);
            stats_kernel<<<NCELL / 256, HDIM, 0, stream>>>(buf2, cni, bg, colsum, colsq, 256);
            long long tot2 = (long long)NCELL * HDIM;
            apply_bn_kernel<<<(unsigned)((tot2 + 255) / 256), 256, 0, stream>>>(
                buf2, cni, bg, colsum, colsq,
                cell_bn_g + layer * HDIM, cell_bn_b + layer * HDIM, buf0, NCELL);
            X = buf0; K = HDIM; Wt = cell_Ws; bg = cell_bs;
        }
    }

    // --- assignment einsum: h_agg[b] = assign[b]^T @ h[b]  (batched WMMA GEMM) ---
    gemm(false, assign, /*sAm=*/1, /*sAk=*/NT_, /*sAb=*/(long long)NC_ * NT_,
         buf0, /*sBb=*/(long long)NC_ * HDIM,
         hagg, /*sCb=*/(long long)NT_ * HDIM,
         NT_, HDIM, NC_, nullptr, nullptr, BATCH);

    // --- concat [h_agg | tissue_feat] ---
    {
        long long totc = (long long)NTIS * (HDIM + FIN_);
        concat_kernel<<<(unsigned)((totc + 255) / 256), 256, 0, stream>>>(hagg, tissue_feat, xt);
    }

    // --- tissue GCN stack ---
    {
        const float* X  = xt;     int K = HDIM + FIN_;
        const float* Wt = tis_W0; const float* bg = tis_b0;
        for (int layer = 0; layer < 3; ++layer) {
            gemm(true, X, K, 1, 0, Wt, 0, tb1, 0, NTIS, HDIM, K, tns, nullptr, 1);
            zero(tb2, (long long)NTIS * HDIM);
            long long tot = (long long)NE_TIS * HDIM;
            scatter_kernel<<<(unsigned)((tot + 255) / 256), 256, 0, stream>>>(
                tis_src, tis_dst, tb1, tb2, tot);
            zero(colsum, HDIM); zero(colsq, HDIM);
            stats_kernel<<<NTIS / 256, HDIM, 0, stream>>>(tb2, tni, bg, colsum, colsq, 256);
            long long tot2 = (long long)NTIS * HDIM;
            apply_bn_kernel<<<(unsigned)((tot2 + 255) / 256), 256, 0, stream>>>(
                tb2, tni, bg, colsum, colsq,
                tis_bn_g + layer * HDIM, tis_bn_b + layer * HDIM, tb0, NTIS);
            X = tb0; K = HDIM; Wt = tis_Ws; bg = tis_bs;
        }
    }

    // --- final linear (+bias) then pool + L2 normalize ---
    gemm(true, tb0, HDIM, 1, 0, lin_W, 0, yv, 0, NTIS, DOUT_, HDIM, nullptr, lin_b, 1);
    pool_norm_kernel<<<BATCH, DOUT_, 0, stream>>>(yv, out);
}